// GraphConvolution_1133871366597
// MI455X (gfx1250) — compile-verified
//
#include <hip/hip_runtime.h>

// GraphConvolution: out[b,v,o] = bias[o] + max_{d<DEG} (h[b] @ W^T)[src[v*DEG+d], o]
// B=16, V=2048, C_IN=C_OUT=128, DEG=32.
//
// Phase 1: support = h @ W^T via native f32 WMMA (v_wmma_f32_16x16x4_f32),
//          W staged in LDS (padded stride to kill bank conflicts).
// Phase 2: wave-per-row gather + segment-max + bias (L2-resident support).

typedef float v2f __attribute__((ext_vector_type(2)));
typedef float v8f __attribute__((ext_vector_type(8)));

#define B_DIM   16
#define V_DIM   2048
#define C_IN    128
#define C_OUT   128
#define DEG     32
#define M_TOTAL (B_DIM * V_DIM)   // 32768 rows (b,v) flattened
#define LDS_STRIDE 132            // 128 + 4 pad: distinct banks for 16-lane b64 reads

// -------- Phase 1: support[m, :] = h[m, :] @ W^T, one wave per 16x128 strip --------
__global__ __launch_bounds__(256) void gcn_gemm_wmma(
    const float* __restrict__ h,       // (M_TOTAL, C_IN)
    const float* __restrict__ W,       // (C_OUT, C_IN)
    float* __restrict__ support)       // (M_TOTAL, C_OUT)
{
    __shared__ float Wlds[C_OUT * LDS_STRIDE];   // 67584 B

    const int t = threadIdx.x;

    // Cooperative copy W (128x128 f32) -> LDS, padded row stride 132.
    // thread t handles row n = t/2, half-row kh = (t&1)*64; both src and dst 16B aligned.
    {
        const int n  = t >> 1;
        const int kh = (t & 1) * 64;
        const float4* src = (const float4*)(W + n * C_IN + kh);
        float4*       dst = (float4*)(&Wlds[n * LDS_STRIDE + kh]);
#pragma unroll
        for (int i = 0; i < 16; ++i) dst[i] = src[i];
    }
    __syncthreads();

    const int wave = t >> 5;
    const int lane = t & 31;
    const int r    = lane & 15;   // A row / B col / D col within the 16x16 tile
    const int half = lane >> 4;   // K pairing half (ISA 16x4 f32 A layout)

    const int    tile = blockIdx.x * 8 + wave;   // m-tile (16 rows)
    const size_t m0   = (size_t)tile * 16;

    v8f acc[8] = {};  // 8 n-tiles of 16 cols each

    const float* arow = h + (m0 + r) * C_IN;

#pragma unroll 4
    for (int k0 = 0; k0 < C_IN; k0 += 4) {
        const int k = k0 + 2 * half;
        // A frag: lanes 0-15 hold K=k0,k0+1; lanes 16-31 hold K=k0+2,k0+3 (rows M=0..15)
        const v2f a = *(const v2f*)(arow + k);
#pragma unroll
        for (int n = 0; n < 8; ++n) {
            // B frag: B[k][n0+c] = W[n0+c][k]; consecutive k -> contiguous float2
            const v2f b = *(const v2f*)(&Wlds[(n * 16 + r) * LDS_STRIDE + k]);
            acc[n] = __builtin_amdgcn_wmma_f32_16x16x4_f32(
                /*neg_a=*/false, a, /*neg_b=*/false, b,
                /*c_mod=*/(short)0, acc[n], /*reuse_a=*/false, /*reuse_b=*/false);
        }
    }

    // D layout: VGPR j -> row m0 + j + 8*half, col n*16 + r
#pragma unroll
    for (int n = 0; n < 8; ++n) {
        float* col = support + (m0 + 8 * half) * C_OUT + n * 16 + r;
#pragma unroll
        for (int j = 0; j < 8; ++j) {
            col[(size_t)j * C_OUT] = acc[n][j];
        }
    }
}

// -------- Phase 2: out[b,v,:] = bias + max over 32 gathered support rows --------
__global__ __launch_bounds__(256) void gcn_gather_max_bias(
    const float* __restrict__ support,  // (M_TOTAL, C_OUT)
    const float* __restrict__ bias,     // (C_OUT)
    const int*   __restrict__ src_idx,  // (V_DIM * DEG)
    float* __restrict__ out)            // (M_TOTAL, C_OUT)
{
    const int wave = threadIdx.x >> 5;
    const int lane = threadIdx.x & 31;
    const int gr   = blockIdx.x * 8 + wave;     // 0..M_TOTAL-1 == b*V + v
    const int b    = gr >> 11;                  // / V_DIM
    const int v    = gr & (V_DIM - 1);

    // Lane l holds src[v*DEG + l]; broadcast across the wave32 via shuffle.
    const int my_src = src_idx[v * DEG + lane];

    const float* base = support + (size_t)b * V_DIM * C_OUT;

    const float ninf = -__builtin_inff();
    float4 m = make_float4(ninf, ninf, ninf, ninf);

#pragma unroll 4
    for (int d = 0; d < DEG; ++d) {
        const int s = __shfl(my_src, d, 32);
        const float4 val = *(const float4*)(base + (size_t)s * C_OUT + lane * 4);
        m.x = fmaxf(m.x, val.x);
        m.y = fmaxf(m.y, val.y);
        m.z = fmaxf(m.z, val.z);
        m.w = fmaxf(m.w, val.w);
    }

    const float4 bi = *(const float4*)(bias + lane * 4);
    float4 res;
    res.x = m.x + bi.x;
    res.y = m.y + bi.y;
    res.z = m.z + bi.z;
    res.w = m.w + bi.w;

    *(float4*)(out + (size_t)gr * C_OUT + lane * 4) = res;
}

extern "C" void kernel_launch(void* const* d_in, const int* in_sizes, int n_in,
                              void* d_out, int out_size, void* d_ws, size_t ws_size,
                              hipStream_t stream) {
    const float* h    = (const float*)d_in[0];   // (16, 2048, 128) f32
    const float* W    = (const float*)d_in[1];   // (128, 128)     f32
    const float* bias = (const float*)d_in[2];   // (128,)         f32
    const int*   src  = (const int*)  d_in[3];   // (65536,)       i32
    // d_in[4] = tgt_idx: implicit (repeat(arange(V), DEG)), unused.

    float* support = (float*)d_ws;               // 16 MB scratch
    float* out     = (float*)d_out;              // (16, 2048, 128) f32

    // Phase 1: 2048 m-tiles, 8 waves/block -> 256 blocks.
    gcn_gemm_wmma<<<M_TOTAL / 16 / 8, 256, 0, stream>>>(h, W, support);

    // Phase 2: 32768 rows, 8 waves/block -> 4096 blocks.
    gcn_gather_max_bias<<<M_TOTAL / 8, 256, 0, stream>>>(support, bias, src, out);
}